// GCMCConv_68049461838610
// MI455X (gfx1250) — compile-verified
//
#include <hip/hip_runtime.h>

typedef float v2f __attribute__((ext_vector_type(2)));
typedef float v8f __attribute__((ext_vector_type(8)));

#define HID 128

// ---------------------------------------------------------------------------
// Kernel 0: zero the scratch region (S accumulator + counts) every call.
// ---------------------------------------------------------------------------
__global__ void gcmc_zero_kernel(float* __restrict__ p, size_t n) {
    size_t i = ((size_t)blockIdx.x * blockDim.x + threadIdx.x) * 4;
    const size_t stride = (size_t)gridDim.x * blockDim.x * 4;
    const float4 z = make_float4(0.f, 0.f, 0.f, 0.f);
    for (; i + 3 < n; i += stride) {
        *(float4*)(p + i) = z;
    }
    if (i < n) {                      // tail (at most 3 floats, one thread)
        for (; i < n; ++i) p[i] = 0.f;
    }
}

// ---------------------------------------------------------------------------
// Kernel 1: per-(dst,rating) feature pre-aggregation.
// One wave32 per edge: coalesced float4 gather of the 128-float source row,
// 4 native f32 atomics per lane into S[d][r][:] (S is L2-resident: 153.6MB
// < 192MB L2), lane 0 counts the edge.
// ---------------------------------------------------------------------------
__global__ __launch_bounds__(256)
void gcmc_edge_scatter_kernel(const float* __restrict__ src_feat,
                              const int* __restrict__ edge_src,
                              const int* __restrict__ edge_dst,
                              const int* __restrict__ rating,
                              float* __restrict__ S,        // [N_DST, NUM_R, HID]
                              float* __restrict__ counts,   // [N_DST]
                              int n_edge, int num_r) {
    const int wv   = threadIdx.x >> 5;
    const int lane = threadIdx.x & 31;
    const int e = blockIdx.x * (blockDim.x >> 5) + wv;
    if (e >= n_edge) return;

    const int s = edge_src[e];
    const int d = edge_dst[e];
    const int r = rating[e];

    const float4 v = ((const float4*)(src_feat + (size_t)s * HID))[lane];
    float* dst = S + ((size_t)d * num_r + r) * HID + lane * 4;
    unsafeAtomicAdd(dst + 0, v.x);
    unsafeAtomicAdd(dst + 1, v.y);
    unsafeAtomicAdd(dst + 2, v.z);
    unsafeAtomicAdd(dst + 3, v.w);
    if (lane == 0) unsafeAtomicAdd(counts + d, 1.0f);
}

// ---------------------------------------------------------------------------
// Kernel 2: fused GEMM pipeline, one 256-thread block (8 waves) per 16 dst
// rows.  Phase A: each wave computes h_neigh[16, 16w:16w+16] =
// (sum_r S_tile @ W_r[r].T) / max(count,1) with v_wmma_f32_16x16x4_f32
// (K = 6*128 = 768 -> 192 WMMA / wave), parks it in padded LDS.
// Phase B: out[16, 16w:16w+16] = relu(dst_feat @ W1.T + h_neigh @ W2.T + b)
// (K = 128 global + 128 LDS -> 64 WMMA / wave).
// EXEC is kept all-ones around all WMMAs (clamped indices, no divergence).
// ---------------------------------------------------------------------------
__global__ __launch_bounds__(256, 2)
void gcmc_gemm_kernel(const float* __restrict__ S,        // [N_DST, NUM_R, HID]
                      const float* __restrict__ counts,   // [N_DST]
                      const float* __restrict__ W_r,      // [NUM_R, HID, HID]
                      const float* __restrict__ dst_feat, // [N_DST, HID]
                      const float* __restrict__ W_lin,    // [HID, 2*HID]
                      const float* __restrict__ b_lin,    // [HID]
                      float* __restrict__ out,            // [N_DST, HID]
                      int n_dst, int num_r) {
    __shared__ float lds_h[16 * 132];   // 16x128 h_neigh tile, padded (+4)

    const int lane = threadIdx.x & 31;
    const int wv   = threadIdx.x >> 5;   // 0..7 : N-tile of 16 channels
    const int ln   = lane & 15;          // lane within half-wave
    const int hi   = lane >> 4;          // half-wave select
    const int m_base = blockIdx.x * 16;  // dst-row tile
    const int n_base = wv * 16;

    // Clamped A-row (keeps loads in range without diverging EXEC).
    int rowA = m_base + ln;
    if (rowA >= n_dst) rowA = n_dst - 1;

    // ---------------- Phase A: h_neigh sums --------------------------------
    v8f acc = {};
    for (int r = 0; r < num_r; ++r) {
        // A(16x4) f32 layout: VGPR0 = K{0|2}, VGPR1 = K{1|3} by lane half.
        const float* Sa = S + ((size_t)rowA * num_r + r) * HID + 2 * hi;
        // B(4x16): B[k][n] = W_r[r][n][k] (row-major W -> contiguous in k).
        const float* Wb = W_r + ((size_t)r * HID + (n_base + ln)) * HID + 2 * hi;
#pragma unroll 8
        for (int k0 = 0; k0 < HID; k0 += 4) {
            v2f a = *(const v2f*)(Sa + k0);
            v2f b = *(const v2f*)(Wb + k0);
            acc = __builtin_amdgcn_wmma_f32_16x16x4_f32(
                false, a, false, b, (short)0, acc, false, false);
        }
    }

    // Mean aggregation + stage tile into LDS.
    // D layout: VGPR j holds (M = j + 8*hi, N = ln).
#pragma unroll
    for (int j = 0; j < 8; ++j) {
        int rr = m_base + 8 * hi + j;
        if (rr >= n_dst) rr = n_dst - 1;
        float c = counts[rr];
        c = (c < 1.0f) ? 1.0f : c;
        lds_h[(j + 8 * hi) * 132 + n_base + ln] = acc[j] / c;
    }
    __syncthreads();

    // ---------------- Phase B: final linear + ReLU -------------------------
    v8f acc2 = {};
    {
        const float* Aa  = dst_feat + (size_t)rowA * HID + 2 * hi;
        const float* Wb1 = W_lin + (size_t)(n_base + ln) * (2 * HID) + 2 * hi;
#pragma unroll 8
        for (int k0 = 0; k0 < HID; k0 += 4) {
            v2f a = *(const v2f*)(Aa + k0);
            v2f b = *(const v2f*)(Wb1 + k0);
            acc2 = __builtin_amdgcn_wmma_f32_16x16x4_f32(
                false, a, false, b, (short)0, acc2, false, false);
        }
        const float* La  = lds_h + ln * 132 + 2 * hi;   // A rows from LDS tile
        const float* Wb2 = Wb1 + HID;                   // W_lin[:, 128:]
#pragma unroll 8
        for (int k0 = 0; k0 < HID; k0 += 4) {
            v2f a = *(const v2f*)(La + k0);
            v2f b = *(const v2f*)(Wb2 + k0);
            acc2 = __builtin_amdgcn_wmma_f32_16x16x4_f32(
                false, a, false, b, (short)0, acc2, false, false);
        }
    }

    const float bias = b_lin[n_base + ln];
#pragma unroll
    for (int j = 0; j < 8; ++j) {
        const int rr = m_base + 8 * hi + j;
        if (rr < n_dst) {
            float v = acc2[j] + bias;
            out[(size_t)rr * HID + n_base + ln] = (v > 0.f) ? v : 0.f;
        }
    }
}

// ---------------------------------------------------------------------------
extern "C" void kernel_launch(void* const* d_in, const int* in_sizes, int n_in,
                              void* d_out, int out_size, void* d_ws, size_t ws_size,
                              hipStream_t stream) {
    (void)n_in; (void)out_size; (void)ws_size;

    const float* src_feat = (const float*)d_in[0];
    const float* dst_feat = (const float*)d_in[1];
    const float* W_r      = (const float*)d_in[2];
    const float* W_lin    = (const float*)d_in[3];
    const float* b_lin    = (const float*)d_in[4];
    const int*   edge_src = (const int*)d_in[5];
    const int*   edge_dst = (const int*)d_in[6];
    const int*   rating   = (const int*)d_in[7];

    const int n_dst  = in_sizes[1] / HID;
    const int n_edge = in_sizes[5];
    const int num_r  = in_sizes[2] / (HID * HID);

    // Workspace layout: S [n_dst * num_r * HID] floats, then counts [n_dst].
    float* S      = (float*)d_ws;
    float* counts = S + (size_t)n_dst * num_r * HID;
    const size_t zfloats = (size_t)n_dst * num_r * HID + (size_t)n_dst;

    gcmc_zero_kernel<<<4096, 256, 0, stream>>>(S, zfloats);

    const int waves_per_blk = 8;
    const int scatter_blocks = (n_edge + waves_per_blk - 1) / waves_per_blk;
    gcmc_edge_scatter_kernel<<<scatter_blocks, 256, 0, stream>>>(
        src_feat, edge_src, edge_dst, rating, S, counts, n_edge, num_r);

    const int gemm_blocks = (n_dst + 15) / 16;
    gcmc_gemm_kernel<<<gemm_blocks, 256, 0, stream>>>(
        S, counts, W_r, dst_feat, W_lin, b_lin, (float*)d_out, n_dst, num_r);
}